// NSMCell_70162585747877
// MI455X (gfx1250) — compile-verified
//
#include <hip/hip_runtime.h>
#include <math.h>

typedef __attribute__((ext_vector_type(2))) float v2f;
typedef __attribute__((ext_vector_type(8))) float v8f;

// ---------- helpers ----------
__device__ __forceinline__ unsigned enc_f32(float f) {
    unsigned u = __float_as_uint(f);
    return (u & 0x80000000u) ? ~u : (u | 0x80000000u);
}
__device__ __forceinline__ float dec_f32(unsigned u) {
    unsigned b = (u & 0x80000000u) ? (u & 0x7FFFFFFFu) : ~u;
    return __uint_as_float(b);
}
__device__ __forceinline__ float elu_fast(float x) {
    // ELU(alpha=1) using the hardware v_exp_f32 path
    return (x > 0.0f) ? x : (__expf(x) - 1.0f);
}

// ---------- init ----------
__global__ void init_kernel(float* rel_logits, unsigned* smax_s, unsigned* smax_r,
                            float* ssum_s, float* ssum_r, int N, int B) {
    int i = blockIdx.x * blockDim.x + threadIdx.x;
    if (i < N) rel_logits[i] = 0.0f;
    if (i < B) {
        smax_s[i] = 0u;   // encoded -inf sentinel (below every real encoding)
        smax_r[i] = 0u;
        ssum_s[i] = 0.0f;
        ssum_r[i] = 0.0f;
    }
}

// ---------- prop_sim = softmax(prop_embeds @ instruction) ----------
__global__ void propsim_kernel(const float* __restrict__ prop_embeds,
                               const float* __restrict__ instr,
                               float* __restrict__ prop_sim) {
    int lane = threadIdx.x;  // 32 threads
    float d[8];
#pragma unroll
    for (int p = 0; p < 8; ++p) {
        float loc = 0.0f;
        for (int i = lane; i < 128; i += 32)
            loc += prop_embeds[p * 128 + i] * instr[i];
#pragma unroll
        for (int m = 16; m >= 1; m >>= 1) loc += __shfl_xor(loc, m, 32);
        d[p] = loc;
    }
    if (lane == 0) {
        float mx = d[0];
#pragma unroll
        for (int p = 1; p < 8; ++p) mx = fmaxf(mx, d[p]);
        float s = 0.0f, e[8];
#pragma unroll
        for (int p = 0; p < 8; ++p) { e[p] = expf(d[p] - mx); s += e[p]; }
#pragma unroll
        for (int p = 0; p < 8; ++p) prop_sim[p] = e[p] / s;
    }
}

// ---------- node path ----------
// state_logits[n] = W_state . elu(instr * sum_p ps[p]*(attrs_p @ Wp^T))
// One wave per 16-node tile; kt-outer / ht-inner with 8 live accumulators so the
// streamed A operand (node_attrs) is loaded exactly once per K step.
// Epilogue fully unrolled: acc[] indices are compile-time constants (no scratch spill).
__global__ __launch_bounds__(128) void node_kernel(
    const float* __restrict__ node_attrs,   // [N,7,128]
    const float* __restrict__ instr,        // [128]
    const float* __restrict__ Wsprop,       // [8,128,128] (use p=0..6)
    const float* __restrict__ W_state,      // [128]
    const float* __restrict__ prop_sim,     // [8]
    float* __restrict__ state_logits,       // [N]
    int N) {
    const int wave = threadIdx.x >> 5;
    const int lane = threadIdx.x & 31;
    const int half = lane >> 4;
    const int mrow = lane & 15;
    const int ntiles = N >> 4;
    const int tile = blockIdx.x * 4 + wave;
    if (tile >= ntiles) return;
    const int nb = tile * 16;

    const float* arow = node_attrs + (size_t)(nb + mrow) * (7 * 128) + 2 * half;

    v8f acc[8];
#pragma unroll
    for (int ht = 0; ht < 8; ++ht)
        acc[ht] = (v8f){0.f, 0.f, 0.f, 0.f, 0.f, 0.f, 0.f, 0.f};

#pragma unroll 1
    for (int p = 0; p < 7; ++p) {
        const float* ap = arow + p * 128;
        // B row base for this lane: B[k][h] = Wp[h][k], h = ht*16 + mrow
        const float* bbase = Wsprop + ((size_t)p * 128 + mrow) * 128 + 2 * half;
        const float psp = prop_sim[p];
#pragma unroll 2
        for (int kt = 0; kt < 32; ++kt) {
            v2f a = *(const v2f*)(ap + kt * 4);  // A[m=mrow][k=4kt+2*half+{0,1}]
            a.x *= psp; a.y *= psp;              // fold prop_sim[p] into A
#pragma unroll
            for (int ht = 0; ht < 8; ++ht) {
                v2f b = *(const v2f*)(bbase + (size_t)ht * (16 * 128) + kt * 4);
                acc[ht] = __builtin_amdgcn_wmma_f32_16x16x4_f32(
                    false, a, false, b, (short)0, acc[ht], false, false);
            }
        }
    }

    float acc_s[8];
#pragma unroll
    for (int v = 0; v < 8; ++v) acc_s[v] = 0.0f;
#pragma unroll
    for (int ht = 0; ht < 8; ++ht) {             // fully unrolled: constant indices
        const int hcol = ht * 16 + mrow;         // this lane's output column (h)
        const float ih = instr[hcol];
        const float wh = W_state[hcol];
#pragma unroll
        for (int v = 0; v < 8; ++v) {            // lane holds D[m=v+8*half][hcol]
            acc_s[v] += elu_fast(acc[ht][v] * ih) * wh;
        }
    }
    // reduce over the 16 h-columns held by the 16 lanes of this half
#pragma unroll
    for (int v = 0; v < 8; ++v) {
        float t = acc_s[v];
#pragma unroll
        for (int m = 1; m <= 8; m <<= 1) t += __shfl_xor(t, m, 32);
        acc_s[v] = t;
    }
    if (mrow == 0) {
#pragma unroll
        for (int v = 0; v < 8; ++v)
            state_logits[nb + half * 8 + v] = acc_s[v];
    }
}

// ---------- edge path ----------
// rel_logits[dst] += dist[src] * (W_relation . elu(instr * (edge_attrs @ W7^T)))
__global__ __launch_bounds__(128) void edge_kernel(
    const float* __restrict__ edge_attrs,   // [E,128]
    const float* __restrict__ instr,        // [128]
    const float* __restrict__ W7,           // Ws_property[7]: [128,128]
    const float* __restrict__ W_rel,        // [128]
    const float* __restrict__ dist,         // [N]
    const int*   __restrict__ edge_src,     // [E]
    const int*   __restrict__ edge_dst,     // [E]
    float* __restrict__ rel_logits,         // [N]
    int E) {
    const int wave = threadIdx.x >> 5;
    const int lane = threadIdx.x & 31;
    const int half = lane >> 4;
    const int mrow = lane & 15;
    const int ntiles = E >> 4;
    const int tile = blockIdx.x * 4 + wave;
    if (tile >= ntiles) return;
    const int eb = tile * 16;

    const float* arow = edge_attrs + (size_t)(eb + mrow) * 128 + 2 * half;
    const float* bbase = W7 + (size_t)mrow * 128 + 2 * half;

    v8f acc[8];
#pragma unroll
    for (int ht = 0; ht < 8; ++ht)
        acc[ht] = (v8f){0.f, 0.f, 0.f, 0.f, 0.f, 0.f, 0.f, 0.f};

#pragma unroll 2
    for (int kt = 0; kt < 32; ++kt) {
        v2f a = *(const v2f*)(arow + kt * 4);
#pragma unroll
        for (int ht = 0; ht < 8; ++ht) {
            v2f b = *(const v2f*)(bbase + (size_t)ht * (16 * 128) + kt * 4);
            acc[ht] = __builtin_amdgcn_wmma_f32_16x16x4_f32(
                false, a, false, b, (short)0, acc[ht], false, false);
        }
    }

    float acc_r[8];
#pragma unroll
    for (int v = 0; v < 8; ++v) acc_r[v] = 0.0f;
#pragma unroll
    for (int ht = 0; ht < 8; ++ht) {             // fully unrolled: constant indices
        const int hcol = ht * 16 + mrow;
        const float ih = instr[hcol];
        const float wh = W_rel[hcol];
#pragma unroll
        for (int v = 0; v < 8; ++v) {
            acc_r[v] += elu_fast(acc[ht][v] * ih) * wh;
        }
    }
#pragma unroll
    for (int v = 0; v < 8; ++v) {
        float t = acc_r[v];
#pragma unroll
        for (int m = 1; m <= 8; m <<= 1) t += __shfl_xor(t, m, 32);
        acc_r[v] = t;
    }
    if (mrow == 0) {
#pragma unroll
        for (int v = 0; v < 8; ++v) {
            int e = eb + half * 8 + v;
            int s = edge_src[e];
            int d = edge_dst[e];
            atomicAdd(&rel_logits[d], dist[s] * acc_r[v]);
        }
    }
}

// ---------- segment softmax passes ----------
__global__ void segmax_kernel(const float* __restrict__ sl, const float* __restrict__ rl,
                              const int* __restrict__ ng,
                              unsigned* smax_s, unsigned* smax_r, int N) {
    int i = blockIdx.x * blockDim.x + threadIdx.x;
    if (i >= N) return;
    int g = ng[i];
    atomicMax(&smax_s[g], enc_f32(sl[i]));
    atomicMax(&smax_r[g], enc_f32(rl[i]));
}

__global__ void segsum_kernel(const float* __restrict__ sl, const float* __restrict__ rl,
                              const int* __restrict__ ng,
                              const unsigned* __restrict__ smax_s, const unsigned* __restrict__ smax_r,
                              float* ssum_s, float* ssum_r, int N) {
    int i = blockIdx.x * blockDim.x + threadIdx.x;
    if (i >= N) return;
    int g = ng[i];
    atomicAdd(&ssum_s[g], __expf(sl[i] - dec_f32(smax_s[g])));
    atomicAdd(&ssum_r[g], __expf(rl[i] - dec_f32(smax_r[g])));
}

__global__ void final_kernel(const float* __restrict__ sl, const float* __restrict__ rl,
                             const int* __restrict__ ng,
                             const unsigned* __restrict__ smax_s, const unsigned* __restrict__ smax_r,
                             const float* __restrict__ ssum_s, const float* __restrict__ ssum_r,
                             const float* __restrict__ prop_sim,
                             float* __restrict__ out, int N) {
    int i = blockIdx.x * blockDim.x + threadIdx.x;
    if (i >= N) return;
    int g = ng[i];
    float prel = prop_sim[7];
    float sm_r = __expf(rl[i] - dec_f32(smax_r[g])) / ssum_r[g];
    float sm_s = __expf(sl[i] - dec_f32(smax_s[g])) / ssum_s[g];
    out[i] = prel * sm_r + (1.0f - prel) * sm_s;
}

// ---------- launch ----------
extern "C" void kernel_launch(void* const* d_in, const int* in_sizes, int n_in,
                              void* d_out, int out_size, void* d_ws, size_t ws_size,
                              hipStream_t stream) {
    const float* node_attrs   = (const float*)d_in[0];
    const float* edge_attrs   = (const float*)d_in[1];
    const float* instruction  = (const float*)d_in[2];
    const float* prop_embeds  = (const float*)d_in[3];
    const float* distribution = (const float*)d_in[4];
    const float* Ws_property  = (const float*)d_in[5];
    const float* W_state      = (const float*)d_in[6];
    const float* W_relation   = (const float*)d_in[7];
    const int*   edge_indices = (const int*)d_in[8];
    const int*   node_graph   = (const int*)d_in[9];

    const int N = in_sizes[4];       // 50000
    const int E = in_sizes[8] / 2;   // 800000
    const int H = 128;
    const int B = 512;               // num_graphs (reference constant)

    float* ws = (float*)d_ws;
    float*    prop_sim     = ws;                          // 8
    float*    state_logits = ws + 8;                      // N
    float*    rel_logits   = ws + 8 + N;                  // N
    unsigned* smax_s       = (unsigned*)(ws + 8 + 2 * N); // B
    unsigned* smax_r       = smax_s + B;                  // B
    float*    ssum_s       = (float*)(smax_r + B);        // B
    float*    ssum_r       = ssum_s + B;                  // B
    float*    out          = (float*)d_out;

    int gridN = (N + 255) / 256;
    init_kernel<<<gridN, 256, 0, stream>>>(rel_logits, smax_s, smax_r, ssum_s, ssum_r, N, B);
    propsim_kernel<<<1, 32, 0, stream>>>(prop_embeds, instruction, prop_sim);

    int ntilesN = N / 16;            // 3125
    node_kernel<<<(ntilesN + 3) / 4, 128, 0, stream>>>(
        node_attrs, instruction, Ws_property, W_state, prop_sim, state_logits, N);

    int ntilesE = E / 16;            // 50000
    edge_kernel<<<(ntilesE + 3) / 4, 128, 0, stream>>>(
        edge_attrs, instruction, Ws_property + 7 * H * H, W_relation, distribution,
        edge_indices, edge_indices + E, rel_logits, E);

    segmax_kernel<<<gridN, 256, 0, stream>>>(state_logits, rel_logits, node_graph, smax_s, smax_r, N);
    segsum_kernel<<<gridN, 256, 0, stream>>>(state_logits, rel_logits, node_graph, smax_s, smax_r,
                                             ssum_s, ssum_r, N);
    final_kernel<<<gridN, 256, 0, stream>>>(state_logits, rel_logits, node_graph, smax_s, smax_r,
                                            ssum_s, ssum_r, prop_sim, out, N);
}